// MultiBoxLoss_17059610099943
// MI455X (gfx1250) — compile-verified
//
#include <hip/hip_runtime.h>
#include <hip/hip_bf16.h>
#include <math.h>

#define Nb 64
#define Mb 24
#define Pp 8732
#define Cc 81
#define THRESH 0.5f

typedef float v2f __attribute__((ext_vector_type(2)));
typedef float v8f __attribute__((ext_vector_type(8)));

__device__ __forceinline__ float sl1(float x) {
    float ax = fabsf(x);
    return (ax < 1.0f) ? 0.5f * ax * ax : ax - 0.5f;
}

// ---------------- kernel 0: zero the accumulators ----------------
__global__ void k0_init(float* acc) {
    if (threadIdx.x < 4) acc[threadIdx.x] = 0.0f;
}

// ---------------- kernel 1: matching + localization loss ----------------
// one block per batch image
__global__ void __launch_bounds__(256)
k1_match(const float* __restrict__ tb,    // [N,M,4] boundary
         const int*   __restrict__ tl,    // [N,M]
         const float* __restrict__ pc,    // [P,4] center form
         const float* __restrict__ ploc,  // [N,P,4]
         int*   __restrict__ lab_ws,      // [N*P] obj idx -> labels
         float* __restrict__ ov_ws,       // [N*P]
         int*   __restrict__ npos_ws,     // [N]
         float* __restrict__ acc)         // [0]=conf [1]=loc [2]=npos
{
    __shared__ float4 sbox[Mb];
    __shared__ float  sarea[Mb];
    __shared__ int    slab[Mb];
    __shared__ int    spf[Mb];
    __shared__ float  sov[Mb][256];
    __shared__ int    spi[Mb][256];
    __shared__ float  rsum[256];
    __shared__ int    risum[256];

    const int n   = blockIdx.x;
    const int tid = threadIdx.x;

    if (tid < Mb) {
        float4 b = ((const float4*)tb)[n * Mb + tid];
        sbox[tid]  = b;
        sarea[tid] = (b.z - b.x) * (b.w - b.y);
        slab[tid]  = tl[n * Mb + tid];
    }
    __syncthreads();

    float bo[Mb];
    int   bp[Mb];
#pragma unroll
    for (int m = 0; m < Mb; ++m) { bo[m] = -1.0f; bp[m] = 0; }

    for (int p = tid; p < Pp; p += 256) {
        float4 pr = ((const float4*)pc)[p];
        float px1 = pr.x - pr.z * 0.5f, py1 = pr.y - pr.w * 0.5f;
        float px2 = pr.x + pr.z * 0.5f, py2 = pr.y + pr.w * 0.5f;
        float areab = pr.z * pr.w;
        float best = -1.0f;
        int   bm   = 0;
#pragma unroll
        for (int m = 0; m < Mb; ++m) {
            float4 a = sbox[m];
            float lx = fmaxf(a.x, px1), ly = fmaxf(a.y, py1);
            float hx = fminf(a.z, px2), hy = fminf(a.w, py2);
            float w  = fmaxf(hx - lx, 0.0f), h = fmaxf(hy - ly, 0.0f);
            float inter = w * h;
            float ov = inter / (sarea[m] + areab - inter);
            if (ov > best)  { best = ov; bm = m; }
            if (ov > bo[m]) { bo[m] = ov; bp[m] = p; }
        }
        lab_ws[n * Pp + p] = bm;
        ov_ws[n * Pp + p]  = best;
    }

#pragma unroll
    for (int m = 0; m < Mb; ++m) { sov[m][tid] = bo[m]; spi[m][tid] = bp[m]; }
    __syncthreads();

    // per-object argmax over P (24 threads scan 256 candidates each)
    if (tid < Mb) {
        float best = -1.0f; int bpp = 0x7fffffff;
        for (int t = 0; t < 256; ++t) {
            float o = sov[tid][t];
            int   q = spi[tid][t];
            if (o > best || (o == best && q < bpp)) { best = o; bpp = q; }
        }
        spf[tid] = bpp;
    }
    __syncthreads();

    // force-assign each object's best prior (in-order, last writer wins)
    if (tid == 0) {
        for (int m = 0; m < Mb; ++m) {
            int pf = spf[m];
            lab_ws[n * Pp + pf] = m;
            ov_ws[n * Pp + pf]  = 1.0f;
        }
    }
    __syncthreads();

    // thresholded labels + localization loss over positives
    int   myNpos = 0;
    float myLoc  = 0.0f;
    for (int p = tid; p < Pp; p += 256) {
        int   obj = lab_ws[n * Pp + p];
        float ov  = ov_ws[n * Pp + p];
        int   lab = (ov < THRESH) ? 0 : slab[obj];
        lab_ws[n * Pp + p] = lab;
        if (lab != 0) {
            ++myNpos;
            float4 a  = sbox[obj];
            float gx = 0.5f * (a.x + a.z), gy = 0.5f * (a.y + a.w);
            float gw = a.z - a.x,          gh = a.w - a.y;
            float4 pr = ((const float4*)pc)[p];
            float tx = (gx - pr.x) * 10.0f / pr.z;
            float ty = (gy - pr.y) * 10.0f / pr.w;
            float tw = __logf(gw / pr.z) * 5.0f;
            float th = __logf(gh / pr.w) * 5.0f;
            float4 pl = ((const float4*)ploc)[n * Pp + p];
            myLoc += sl1(pl.x - tx) + sl1(pl.y - ty) + sl1(pl.z - tw) + sl1(pl.w - th);
        }
    }
    rsum[tid]  = myLoc;
    risum[tid] = myNpos;
    __syncthreads();
    for (int s = 128; s > 0; s >>= 1) {
        if (tid < s) { rsum[tid] += rsum[tid + s]; risum[tid] += risum[tid + s]; }
        __syncthreads();
    }
    if (tid == 0) {
        npos_ws[n] = risum[0];
        atomicAdd(&acc[1], rsum[0]);
        atomicAdd(&acc[2], (float)risum[0]);
    }
}

// ---------------- kernel 2: confidence loss via WMMA logsumexp ----------------
// Each wave32 owns a 16-prior tile. The 16x81 tile is staged through LDS with a
// single coalesced global pass; sum-of-exp over classes is D = A x ones via
// V_WMMA_F32_16X16X4_F32 accumulated across 21 K-chunks of 4.
#define TPB_TILES ((Pp + 15) / 16)   // 546
#define TILE_FLTS (16 * Cc)          // 1296
#define TILE_PAD  1312               // 41 * 32, per-wave LDS span

__global__ void __launch_bounds__(256)
k2_conf(const float* __restrict__ scores, // [N,P,C]
        const int*   __restrict__ labels, // [N*P]
        const int*   __restrict__ npos,   // [N]
        float*       __restrict__ acc)
{
    __shared__ float stile[8][TILE_PAD];

    const int wid  = threadIdx.x >> 5;
    const int lane = threadIdx.x & 31;
    const int tile = blockIdx.x * 8 + wid;
    if (tile >= Nb * TPB_TILES) return;          // grid is exact; wave-uniform
    const int n     = tile / TPB_TILES;
    const int pbase = (tile - n * TPB_TILES) * 16;
    const int nrows = min(16, Pp - pbase);
    const int vs    = nrows * Cc;                // valid contiguous span
    const float* span = scores + (size_t)(n * Pp + pbase) * Cc;
    __builtin_prefetch(span, 0, 1);              // global_prefetch_b8

    // coalesced, unconditional (index-clamped) global -> LDS tile copy
    float* lds = stile[wid];
#pragma unroll
    for (int i = lane; i < TILE_PAD; i += 32) {
        int gi = (i < vs) ? i : (vs - 1);
        lds[i] = span[gi];
    }
    __syncthreads();

    const int row  = lane & 15;
    const int hi   = lane >> 4;                  // 0: K=0,1   1: K=2,3 (A layout)
    const int roff = row * Cc;

    // pass 1: row max from LDS (index clamp instead of predication)
    float mx = -INFINITY;
#pragma unroll
    for (int j = 0; j < 21; ++j) {
        int c0 = 4 * j + 2 * hi;
        int ca = (c0     < Cc) ? c0     : (Cc - 1);
        int cb = (c0 + 1 < Cc) ? c0 + 1 : (Cc - 1);
        mx = fmaxf(mx, fmaxf(lds[roff + ca], lds[roff + cb]));
    }
    mx = fmaxf(mx, __shfl_xor(mx, 16, 32));

    // pass 2: sum of exp via WMMA with ones B-matrix, f32 accumulate
    v2f ones; ones.x = 1.0f; ones.y = 1.0f;
    v8f accv = {0.f, 0.f, 0.f, 0.f, 0.f, 0.f, 0.f, 0.f};
#pragma unroll
    for (int j = 0; j < 21; ++j) {
        int c0 = 4 * j + 2 * hi;
        int ca = (c0     < Cc) ? c0     : (Cc - 1);
        int cb = (c0 + 1 < Cc) ? c0 + 1 : (Cc - 1);
        float e0 = __expf(lds[roff + ca] - mx);  // always real data: exp <= 1
        float e1 = __expf(lds[roff + cb] - mx);
        v2f a;
        a.x = (c0     < Cc) ? e0 : 0.0f;         // folds away for j < 20
        a.y = (c0 + 1 < Cc) ? e1 : 0.0f;
        accv = __builtin_amdgcn_wmma_f32_16x16x4_f32(
            false, a, false, ones, (short)0, accv, false, false);
    }

    // extraction: column N=0 lives in lanes 0 (rows 0..7) and 16 (rows 8..15)
    float local = 0.0f;
    const int k = 3 * npos[n];
#pragma unroll
    for (int r = 0; r < 8; ++r) {
        float mxr = __shfl(mx, (hi ? 8 : 0) + r, 32);   // all lanes participate
        float se  = accv[r];
        if (lane == 0 || lane == 16) {
            int rr = hi * 8 + r;
            int pp = pbase + rr;
            if (pp < Pp) {
                int   lab  = labels[n * Pp + pp];
                float ts   = lds[rr * Cc + lab];         // score[label] from LDS
                float conf = mxr + __logf(se) - ts;
                float w    = (lab != 0 ? 1.0f : 0.0f) + (pp < k ? 1.0f : 0.0f);
                local += conf * w;
            }
        }
    }
    float other = __shfl(local, 16, 32);
    if (lane == 0) atomicAdd(&acc[0], local + other);
}

// ---------------- kernel 3: finalize ----------------
__global__ void k3_final(const float* __restrict__ acc, float* __restrict__ out) {
    float npt = acc[2];
    out[0] = acc[0] / npt + acc[1] / (4.0f * npt);
}

extern "C" void kernel_launch(void* const* d_in, const int* in_sizes, int n_in,
                              void* d_out, int out_size, void* d_ws, size_t ws_size,
                              hipStream_t stream) {
    const float* predictedLocs        = (const float*)d_in[0];
    const float* predictedClassScores = (const float*)d_in[1];
    const float* trueBboxes           = (const float*)d_in[2];
    const int*   trueLabels           = (const int*)d_in[3];
    const float* priorsCenter         = (const float*)d_in[4];
    float* out = (float*)d_out;

    char* ws = (char*)d_ws;
    float* acc     = (float*)ws;                                  // 4 floats
    int*   lab_ws  = (int*)(ws + 64);                             // N*P ints
    float* ov_ws   = (float*)(ws + 64 + (size_t)Nb * Pp * 4);     // N*P floats
    int*   npos_ws = (int*)(ws + 64 + (size_t)Nb * Pp * 8);       // N ints

    k0_init<<<1, 32, 0, stream>>>(acc);
    k1_match<<<Nb, 256, 0, stream>>>(trueBboxes, trueLabels, priorsCenter,
                                     predictedLocs, lab_ws, ov_ws, npos_ws, acc);
    int tiles  = Nb * TPB_TILES;
    int blocks = (tiles + 7) / 8;      // 8 waves per 256-thread block, exact
    k2_conf<<<blocks, 256, 0, stream>>>(predictedClassScores, lab_ws, npos_ws, acc);
    k3_final<<<1, 1, 0, stream>>>(acc, out);
}